// PatternMatchingLoss_3599182594548
// MI455X (gfx1250) — compile-verified
//
#include <hip/hip_runtime.h>
#include <hip/hip_bf16.h>
#include <math.h>

#define SZ   512
#define S1   513
#define NB   64
#define NH   30
#define CIN  50   // 20 + NH

typedef __attribute__((ext_vector_type(2))) float v2f;
typedef __attribute__((ext_vector_type(8))) float v8f;

// ---------------------------------------------------------------------------
// Kernel 1: per-batch gap mask (argmax channel != 0), stable left-pack into
// hmm[B,50,512], token count ls[b]. One 512-thread block per batch.
// ---------------------------------------------------------------------------
__global__ __launch_bounds__(512) void build_hmm_kernel(
    const float* __restrict__ x,        // [B,21,512]
    const float* __restrict__ seq_hmm,  // [30,512]
    float* __restrict__ hmm,            // [B,50,512]
    int* __restrict__ ls)               // [B]
{
    __shared__ int scan_s[512];
    const int b = blockIdx.x, l = threadIdx.x;
    const float* xb = x + (size_t)b * 21 * SZ;

    // argmax over 21 channels, first-max wins (strict >)
    float best = xb[l];
    int idx = 0;
    #pragma unroll
    for (int c = 1; c < 21; ++c) {
        float v = xb[(size_t)c * SZ + l];
        if (v > best) { best = v; idx = c; }
    }
    const int m = (idx != 0) ? 1 : 0;
    scan_s[l] = m;
    __syncthreads();
    // inclusive Hillis-Steele scan
    for (int off = 1; off < 512; off <<= 1) {
        int v = 0;
        if (l >= off) v = scan_s[l - off];
        __syncthreads();
        scan_s[l] += v;
        __syncthreads();
    }
    const int total = scan_s[511];
    const int dest  = scan_s[l] - m;  // exclusive prefix = packed position

    float* hb = hmm + (size_t)b * CIN * SZ;
    if (m) {
        #pragma unroll
        for (int c = 0; c < 20; ++c)
            hb[(size_t)c * SZ + dest] = xb[(size_t)(c + 1) * SZ + l];
        #pragma unroll
        for (int h = 0; h < NH; ++h)
            hb[(size_t)(20 + h) * SZ + dest] = seq_hmm[(size_t)h * SZ + l];
    }
    if (l >= total) {  // zero the invalid (masked) tail columns
        #pragma unroll
        for (int c = 0; c < CIN; ++c) hb[(size_t)c * SZ + l] = 0.f;
    }
    if (l == 0) ls[b] = total;
}

// ---------------------------------------------------------------------------
// Kernel 2: ss3 head GEMM via fp32 WMMA (exact): logits[b,3,512] =
//   W[3,50] x hmm[b,50,512] + bias.  One wave per 16-column tile.
// Padding via 0/1 mask multiplies on unconditionally-loaded (clamped-address)
// values -> no exec-mask branches; boundary K-chunk peeled out of the loop.
// ---------------------------------------------------------------------------
__global__ __launch_bounds__(128) void ss3_wmma_kernel(
    const float* __restrict__ hmm,   // [B,50,512]
    const float* __restrict__ W,     // [3,50]
    const float* __restrict__ bias,  // [3]
    float* __restrict__ logits)      // [B,3,512]
{
    const int wid  = blockIdx.x * 4 + (threadIdx.x >> 5);  // 0 .. B*32-1
    const int lane = threadIdx.x & 31;
    const int b    = wid >> 5;
    const int l0   = (wid & 31) << 4;

    const int m  = lane & 15;   // A: row M; B: column N
    const int hi = lane >> 4;   // 0 -> K {0,1}, 1 -> K {2,3} within a 4-chunk
    const int mc = (m < 3) ? m : 2;          // clamped row (always-valid addr)
    const float am = (m < 3) ? 1.f : 0.f;    // A row pad mask
    const float* Wr = W + mc * CIN;
    const float* hb = hmm + (size_t)b * CIN * SZ + l0 + m;

    v8f acc = {};
    // interior chunks: K = kk*4 .. kk*4+3, all columns < 48 < CIN (no pad)
    #pragma unroll
    for (int kk = 0; kk < 12; ++kk) {
        const int ka = kk * 4 + (hi ? 2 : 0);
        v2f av, bv;
        // A (16x4): lane m<16 holds K=0,1 ; lane m+16 holds K=2,3
        av.x = am * Wr[ka];
        av.y = am * Wr[ka + 1];
        // B (4x16): v0 lanes0-15 = row K0, lanes16-31 = row K2; v1 = K1/K3
        bv.x = hb[(size_t)ka * SZ];
        bv.y = hb[(size_t)(ka + 1) * SZ];
        acc = __builtin_amdgcn_wmma_f32_16x16x4_f32(
            false, av, false, bv, (short)0, acc, false, false);
    }
    // boundary chunk: K = 48..51; cols 50,51 (hi lanes) are zero padding
    {
        const float sa = hi ? 0.f : am;   // A pad: row mask AND col<50
        const float sb = hi ? 0.f : 1.f;  // B pad: col<50
        const int ka = 48;                // clamped address for hi lanes too
        v2f av, bv;
        av.x = sa * Wr[ka];
        av.y = sa * Wr[ka + 1];
        bv.x = sb * hb[(size_t)ka * SZ];
        bv.y = sb * hb[(size_t)(ka + 1) * SZ];
        acc = __builtin_amdgcn_wmma_f32_16x16x4_f32(
            false, av, false, bv, (short)0, acc, false, false);
    }
    // D: VGPR r, lanes 0-15 hold M=r -> rows 0..2 are acc[0..2] on lanes 0-15
    if (lane < 16) {
        float* lg = logits + (size_t)b * 3 * SZ + l0 + lane;
        lg[0]      = acc[0] + bias[0];
        lg[SZ]     = acc[1] + bias[1];
        lg[2 * SZ] = acc[2] + bias[2];
    }
}

// ---------------------------------------------------------------------------
// Kernel 3: per-position 3-class log-softmax, then cumsum over L -> cs[B,3,513]
// ---------------------------------------------------------------------------
__global__ __launch_bounds__(512) void softmax_cumsum_kernel(
    const float* __restrict__ logits,  // [B,3,512]
    float* __restrict__ cs)            // [B,3,513]
{
    __shared__ float s0[512], s1[512], s2[512];
    const int b = blockIdx.x, l = threadIdx.x;
    const float* lg = logits + (size_t)b * 3 * SZ + l;
    float a0 = lg[0], a1 = lg[SZ], a2 = lg[2 * SZ];
    float mx  = fmaxf(a0, fmaxf(a1, a2));
    float lse = mx + __logf(__expf(a0 - mx) + __expf(a1 - mx) + __expf(a2 - mx));
    s0[l] = a0 - lse; s1[l] = a1 - lse; s2[l] = a2 - lse;
    __syncthreads();
    for (int off = 1; off < 512; off <<= 1) {
        float v0 = 0, v1 = 0, v2 = 0;
        if (l >= off) { v0 = s0[l - off]; v1 = s1[l - off]; v2 = s2[l - off]; }
        __syncthreads();
        if (l >= off) { s0[l] += v0; s1[l] += v1; s2[l] += v2; }
        __syncthreads();
    }
    float* csb = cs + (size_t)b * 3 * S1;
    if (l == 0) { csb[0] = 0.f; csb[S1] = 0.f; csb[2 * S1] = 0.f; }
    csb[l + 1]          = s0[l];
    csb[S1 + l + 1]     = s1[l];
    csb[2 * S1 + l + 1] = s2[l];
}

// ---------------------------------------------------------------------------
// Kernel 4: R = exp(Q) + 1e-8  (3*513*513, stays L2-resident afterwards)
// ---------------------------------------------------------------------------
__global__ __launch_bounds__(256) void expQ_kernel(
    const float* __restrict__ Q, float* __restrict__ R, int n)
{
    int i = blockIdx.x * 256 + threadIdx.x;
    if (i < n) R[i] = __expf(Q[i]) + 1e-8f;
}

// ---------------------------------------------------------------------------
// Kernel 5: 24-step HMM forward recursion in normalized linear space.
// One 512-thread block (16 wave32) per batch; runs all steps with barriers.
// a'[k] = sum_{j<k} (exp(cs_k-cs_j)+1e-8) R[j,k] a[j]
//       + sum_{k<=j<512} 1e-8 R[j,k] a[j] + (1+1e-8) R[512,k] a[512]
// Per-step normalization == the reference's log-space LSE normalization.
// a and cs are packed as float2 in LDS (one ds_load_b64 per inner iteration).
// ---------------------------------------------------------------------------
__global__ __launch_bounds__(512) void hmm_forward_kernel(
    const float* __restrict__ cs,       // [B,3,513]
    const float* __restrict__ R,        // [3,513,513]
    const int* __restrict__ ls,         // [B]
    const int* __restrict__ pattern,    // [T]
    float* __restrict__ out, int T)
{
    __shared__ float2 ac_s[S1];        // .x = a (normalized alpha), .y = cs
    __shared__ float part[2][S1];
    __shared__ float red[16];          // one partial per wave

    const int b    = blockIdx.x;
    const int t0   = threadIdx.x;
    const int lane = t0 & 31;
    const int wv   = t0 >> 5;
    const int half = t0 >> 8;          // j-range half: [0,256) or [256,512)
    const int kt   = t0 & 255;
    const int jlo  = half ? 256 : 0;
    const int jhi  = half ? 512 : 256; // j==512 row handled explicitly

    for (int k = t0; k < S1; k += 512) ac_s[k].x = (k == 0) ? 1.f : 0.f;
    float n_acc = 0.f;
    __syncthreads();

    for (int t = 0; t < T; ++t) {
        const int c = pattern[t];                       // uniform scalar load
        const float* csb = cs + ((size_t)b * 3 + c) * S1;
        for (int k = t0; k < S1; k += 512) ac_s[k].y = csb[k];
        __syncthreads();

        const float* Rc = R + (size_t)c * S1 * S1;
        // ---- phase 1: per-half partial contractions over j ----
        for (int kk = 0; kk < 3; ++kk) {
            int k;
            if (kk == 0)      k = kt;
            else if (kk == 1) k = kt + 256;
            else { if (kt != 0) break; k = 512; }
            const float csk = ac_s[k].y;
            const float* Rk = Rc + k;
            const int je = (k < jhi) ? ((k > jlo) ? k : jlo) : jhi;
            float acc = 0.f, tail = 0.f;
            int j = jlo;
            #pragma unroll 4
            for (; j < je; ++j) {                 // exp region (j < k)
                float2 ac = ac_s[j];
                acc += (__expf(csk - ac.y) + 1e-8f) * Rk[(size_t)j * S1] * ac.x;
            }
            #pragma unroll 4
            for (; j < jhi; ++j)                  // 1e-8 region (k <= j < 512)
                tail += Rk[(size_t)j * S1] * ac_s[j].x;
            acc += 1e-8f * tail;
            if (half == 1)                        // j == 512 row: weight (1+1e-8)
                acc += (1.f + 1e-8f) * Rk[(size_t)512 * S1] * ac_s[512].x;
            part[half][k] = acc;
        }
        __syncthreads();

        // ---- phase 2: combine halves, wave-level reduce, normalize ----
        float v0 = part[0][t0] + part[1][t0];
        float v1 = 0.f;
        float lsum = v0;
        if (t0 == 0) { v1 = part[0][512] + part[1][512]; lsum += v1; }
        #pragma unroll
        for (int off = 16; off > 0; off >>= 1)    // wave32 shuffle reduction
            lsum += __shfl_xor(lsum, off, 32);
        if (lane == 0) red[wv] = lsum;
        __syncthreads();
        float total = 0.f;
        #pragma unroll
        for (int w = 0; w < 16; ++w) total += red[w];
        const float inv = 1.f / total;
        ac_s[t0].x = v0 * inv;
        if (t0 == 0) ac_s[512].x = v1 * inv;
        n_acc += __logf(total);
        __syncthreads();
    }

    if (t0 == 0) {
        const int L = ls[b];
        out[b] = (__logf(ac_s[L].x) + n_acc) / (float)L;
    }
}

// ---------------------------------------------------------------------------
extern "C" void kernel_launch(void* const* d_in, const int* in_sizes, int n_in,
                              void* d_out, int out_size, void* d_ws, size_t ws_size,
                              hipStream_t stream)
{
    const float* x       = (const float*)d_in[0];   // [64,21,512]
    const float* seq_hmm = (const float*)d_in[1];   // [30,512]
    const float* W       = (const float*)d_in[2];   // [3,50]
    const float* bias    = (const float*)d_in[3];   // [3]
    const float* Q       = (const float*)d_in[4];   // [3,513,513]
    const int*   pattern = (const int*)d_in[5];     // [T]
    const int T = in_sizes[5];
    float* out = (float*)d_out;

    char* ws = (char*)d_ws;
    const size_t off_hmm    = 0;
    const size_t off_logits = off_hmm    + (size_t)NB * CIN * SZ * 4;   // 6,553,600
    const size_t off_cs     = off_logits + (size_t)NB * 3 * SZ * 4;     // +393,216
    const size_t off_R      = off_cs     + (size_t)NB * 3 * S1 * 4;     // +393,984
    const size_t off_ls     = off_R      + (size_t)3 * S1 * S1 * 4;     // +3,158,028
    float* hmm    = (float*)(ws + off_hmm);
    float* logits = (float*)(ws + off_logits);
    float* cs     = (float*)(ws + off_cs);
    float* R      = (float*)(ws + off_R);
    int*   lsbuf  = (int*)  (ws + off_ls);

    build_hmm_kernel<<<NB, 512, 0, stream>>>(x, seq_hmm, hmm, lsbuf);
    ss3_wmma_kernel<<<NB * 32 / 4, 128, 0, stream>>>(hmm, W, bias, logits);
    softmax_cumsum_kernel<<<NB, 512, 0, stream>>>(logits, cs);
    const int nR = 3 * S1 * S1;
    expQ_kernel<<<(nR + 255) / 256, 256, 0, stream>>>(Q, R, nR);
    hmm_forward_kernel<<<NB, 512, 0, stream>>>(cs, R, lsbuf, pattern, out, T);
}